// DifferentiableSATSolver_18571438588593
// MI455X (gfx1250) — compile-verified
//
#include <hip/hip_runtime.h>

#define B_    64
#define N_    10000
#define C_    50000
#define L_    3
#define ITERS 100
#define STEP  0.1f
#define TPB   1024

// Prep: pack each literal as int2 { idx | (sign>0)<<31 , bits(sign*w[idx]) }.
// One coalesced b64 load per literal in the hot loop instead of 3 dword loads.
__global__ __launch_bounds__(256) void sat_prep(const float* __restrict__ w,
                                                const int* __restrict__ cvar,
                                                const float* __restrict__ csign,
                                                int2* __restrict__ pl) {
    int i = blockIdx.x * blockDim.x + threadIdx.x;
    if (i < C_ * L_) {
        const int   v = cvar[i];
        const float g = csign[i];
        int2 p;
        p.x = (int)((unsigned)v | (g > 0.0f ? 0x80000000u : 0u));
        p.y = __float_as_int(g * w[v]);
        pl[i] = p;
    }
}

// Address-space-qualified pointer types matching the async-to-LDS builtin's
// prototype (AS1 int* source, AS3 int* LDS destination).
typedef __attribute__((address_space(1))) int* gbl_i32p;
typedef __attribute__((address_space(3))) int* lds_i32p;

__device__ __forceinline__ void async_copy_b32(const float* gsrc, float* ldst) {
#if __has_builtin(__builtin_amdgcn_global_load_async_to_lds_b32)
    int* g_generic = (int*)gsrc;   // const_cast + reinterpret (generic AS)
    int* l_generic = (int*)ldst;
    __builtin_amdgcn_global_load_async_to_lds_b32(
        (gbl_i32p)g_generic, (lds_i32p)l_generic, 0, 0);
#else
    *ldst = *gsrc;
#endif
}

__device__ __forceinline__ void wait_async_done() {
#if __has_builtin(__builtin_amdgcn_global_load_async_to_lds_b32) && \
    __has_builtin(__builtin_amdgcn_s_wait_asynccnt)
    __builtin_amdgcn_s_wait_asynccnt(0);
#endif
}

__global__ __launch_bounds__(TPB) void sat_solver(const float* __restrict__ a0,
                                                  const int2* __restrict__ pl,
                                                  float* __restrict__ a_out,
                                                  float* __restrict__ sat_out) {
    __shared__ float s_a[N_];      // 40 KB: per-batch assignment (read copy)
    __shared__ float s_delta[N_];  // 40 KB: per-iteration scatter-add accumulator

    const int b   = blockIdx.x;
    const int tid = threadIdx.x;
    const float* __restrict__ ab = a0 + (size_t)b * N_;

    // ---- Stage this batch's assignment row into LDS (async, ASYNCcnt) ----
    for (int i = tid; i < N_; i += TPB) {
        async_copy_b32(ab + i, s_a + i);
    }
    for (int i = tid; i < N_; i += TPB) s_delta[i] = 0.0f;
    wait_async_done();
    __syncthreads();

    // ---- 100 bulk-synchronous solver iterations ----
    for (int it = 0; it < ITERS; ++it) {
        // Phase 1: clause evaluation + scatter-add into s_delta (ds_add_f32)
        for (int c = tid; c < C_; c += TPB) {
            const int base = 3 * c;
            const int2 p0 = pl[base + 0];
            const int2 p1 = pl[base + 1];
            const int2 p2 = pl[base + 2];
            const int  i0 = p0.x & 0x7fffffff;
            const int  i1 = p1.x & 0x7fffffff;
            const int  i2 = p2.x & 0x7fffffff;
            const float v0 = s_a[i0];
            const float v1 = s_a[i1];
            const float v2 = s_a[i2];
            const float l0 = (p0.x < 0) ? v0 : 1.0f - v0;   // MSB set => sign>0
            const float l1 = (p1.x < 0) ? v1 : 1.0f - v1;
            const float l2 = (p2.x < 0) ? v2 : 1.0f - v2;
            const float sat = fmaxf(fmaxf(fmaxf(l0, l1), l2), 0.0f);
            const float f   = STEP * (1.0f - sat);           // STEP * viol
            atomicAdd(&s_delta[i0], f * __int_as_float(p0.y));
            atomicAdd(&s_delta[i1], f * __int_as_float(p1.y));
            atomicAdd(&s_delta[i2], f * __int_as_float(p2.y));
        }
        __syncthreads();
        // Phase 2: apply + clip, reset accumulator
        for (int i = tid; i < N_; i += TPB) {
            const float na = s_a[i] + s_delta[i];
            s_a[i]     = fminf(fmaxf(na, 0.0f), 1.0f);
            s_delta[i] = 0.0f;
        }
        __syncthreads();
    }

    // ---- Write final assignment ----
    float* __restrict__ aob = a_out + (size_t)b * N_;
    for (int i = tid; i < N_; i += TPB) aob[i] = s_a[i];
    __syncthreads();

    // ---- satisfaction[b] = prod over clauses of final sat ----
    float p = 1.0f;
    for (int c = tid; c < C_; c += TPB) {
        const int base = 3 * c;
        const int2 p0 = pl[base + 0];
        const int2 p1 = pl[base + 1];
        const int2 p2 = pl[base + 2];
        const float v0 = s_a[p0.x & 0x7fffffff];
        const float v1 = s_a[p1.x & 0x7fffffff];
        const float v2 = s_a[p2.x & 0x7fffffff];
        const float l0 = (p0.x < 0) ? v0 : 1.0f - v0;
        const float l1 = (p1.x < 0) ? v1 : 1.0f - v1;
        const float l2 = (p2.x < 0) ? v2 : 1.0f - v2;
        p *= fmaxf(fmaxf(fmaxf(l0, l1), l2), 0.0f);
    }
    s_delta[tid] = p;
    __syncthreads();
    for (int s = TPB / 2; s > 0; s >>= 1) {
        if (tid < s) s_delta[tid] *= s_delta[tid + s];
        __syncthreads();
    }
    if (tid == 0) sat_out[b] = s_delta[0];
}

extern "C" void kernel_launch(void* const* d_in, const int* in_sizes, int n_in,
                              void* d_out, int out_size, void* d_ws, size_t ws_size,
                              hipStream_t stream) {
    const float* a0 = (const float*)d_in[0];  // (B, N) f32
    const float* w  = (const float*)d_in[1];  // (N,)   f32
    const int*   cv = (const int*)  d_in[2];  // (C, L) i32
    const float* cs = (const float*)d_in[3];  // (C, L) f32

    float* a_out   = (float*)d_out;                 // (B, N)
    float* sat_out = a_out + (size_t)B_ * N_;       // (B,)
    int2*  pl      = (int2*)d_ws;                   // (C*L) packed literals

    sat_prep<<<(C_ * L_ + 255) / 256, 256, 0, stream>>>(w, cv, cs, pl);
    sat_solver<<<B_, TPB, 0, stream>>>(a0, pl, a_out, sat_out);
}